// TokenMixing_27900107555053
// MI455X (gfx1250) — compile-verified
//
#include <hip/hip_runtime.h>
#include <hip/hip_bf16.h>

// Sizes (compile-time constants from the reference)
#define BSZ 4
#define SEQ 512
#define DIM 256
#define KAD 64

typedef __attribute__((ext_vector_type(2))) float v2f;
typedef __attribute__((ext_vector_type(8))) float v8f;
typedef __attribute__((ext_vector_type(4))) unsigned int u32x4;
typedef __attribute__((ext_vector_type(8))) int i32x8;
typedef __attribute__((ext_vector_type(4))) int i32x4;

// ---------------------------------------------------------------------------
// Kernel 1: W[b][o][i] = weight[o][i] + adapt[b] . aw_w[o*D+i] + aw_b[o*D+i]
// Streams the 16.8 MB aw_w exactly once, fully coalesced. Memory-bound floor.
// ---------------------------------------------------------------------------
__global__ void build_W_kernel(const float* __restrict__ adapt,   // [B,K]
                               const float* __restrict__ weight,  // [D*D]
                               const float* __restrict__ aw_w,    // [D*D, K]
                               const float* __restrict__ aw_b,    // [D*D]
                               float* __restrict__ Wws)           // [B, D*D]
{
    __shared__ float s_adapt[BSZ * KAD];
    int t = threadIdx.x;
    s_adapt[t] = adapt[t];
    __syncthreads();

    int row = blockIdx.x * 256 + t;      // 0 .. 65535
    const float4* wr = (const float4*)(aw_w + (size_t)row * KAD);

    float acc[BSZ] = {0.f, 0.f, 0.f, 0.f};
#pragma unroll
    for (int kv = 0; kv < KAD / 4; ++kv) {
        float4 v = wr[kv];
#pragma unroll
        for (int bb = 0; bb < BSZ; ++bb) {
            const float* ad = &s_adapt[bb * KAD + kv * 4];
            acc[bb] += v.x * ad[0] + v.y * ad[1] + v.z * ad[2] + v.w * ad[3];
        }
    }

    float base = weight[row] + aw_b[row];
#pragma unroll
    for (int bb = 0; bb < BSZ; ++bb)
        Wws[(size_t)bb * (DIM * DIM) + row] = base + acc[bb];
}

// ---------------------------------------------------------------------------
// Kernel 2: ab[b][o] = adapt[b] . ab_w[o] + ab_b[o]   (tiny)
// ---------------------------------------------------------------------------
__global__ void build_ab_kernel(const float* __restrict__ adapt, // [B,K]
                                const float* __restrict__ ab_w,  // [D,K]
                                const float* __restrict__ ab_b,  // [D]
                                float* __restrict__ abws)        // [B,D]
{
    __shared__ float s_adapt[BSZ * KAD];
    int o = threadIdx.x;
    s_adapt[o] = adapt[o];
    __syncthreads();

    const float* wr = ab_w + (size_t)o * KAD;
    float bias = ab_b[o];
    float acc[BSZ] = {0.f, 0.f, 0.f, 0.f};
#pragma unroll 8
    for (int k = 0; k < KAD; ++k) {
        float w = wr[k];
#pragma unroll
        for (int bb = 0; bb < BSZ; ++bb) acc[bb] += w * s_adapt[bb * KAD + k];
    }
#pragma unroll
    for (int bb = 0; bb < BSZ; ++bb) abws[bb * DIM + o] = bias + acc[bb];
}

// ---------------------------------------------------------------------------
// Tensor Data Mover: 2D tile (tile_d1 rows x 32 dwords) from global to LDS,
// row stride `stride_d0` dwords in memory, LDS rows padded 32 -> 36 dwords
// (pad_enable=1, pad_interval=4 => every 32 dwords, pad_amount=3 => 4 dwords).
// tensor dims == tile dims, so OOB logic is inert. data_size=2 => 4 bytes.
// D# layout per CDNA5 ISA ch.8 (group0: count/lds/global/type; group1: dims).
// ---------------------------------------------------------------------------
__device__ __forceinline__ void tdm_load_tile_2d(unsigned lds_byte,
                                                 const float* gptr,
                                                 unsigned tile_d0,   // dwords per row (32)
                                                 unsigned tile_d1,   // rows
                                                 unsigned stride_d0) // dwords
{
    unsigned long long ga = (unsigned long long)(uintptr_t)gptr;
    u32x4 g0;
    g0[0] = 1u;                                     // count=1, user descriptor
    g0[1] = lds_byte;                               // lds_addr (bytes)
    g0[2] = (unsigned)(ga & 0xFFFFFFFFu);           // global_addr[31:0]
    g0[3] = (unsigned)((ga >> 32) & 0x01FFFFFFu)    // global_addr[56:32]
          | (2u << 30);                             // type=2 ("image")
    i32x8 g1;
    g1[0] = (int)((2u << 16)        // data_size = 4 bytes
                | (1u << 20)        // pad_enable
                | (4u << 22)        // pad_interval: 2^(4+1)=32 dwords
                | (3u << 25));      // pad_amount: 3+1 = 4 dwords
    g1[1] = (int)((tile_d0 & 0xFFFFu) << 16);             // tensor_dim0[15:0]
    g1[2] = (int)((tile_d0 >> 16) | ((tile_d1 & 0xFFFFu) << 16)); // td0 hi | td1 lo
    g1[3] = (int)((tile_d1 >> 16) | (tile_d0 << 16));     // td1 hi | tile_dim0
    g1[4] = (int)(tile_d1 & 0xFFFFu);                     // tile_dim1 (tile_dim2=0)
    g1[5] = (int)stride_d0;                               // tensor_dim0_stride[31:0]
    g1[6] = 0;                                            // stride hi / dim1_stride lo
    g1[7] = 0;
    i32x4 z4 = {0, 0, 0, 0};
#if __clang_major__ >= 23
    i32x8 z8 = {0, 0, 0, 0, 0, 0, 0, 0};
    __builtin_amdgcn_tensor_load_to_lds(g0, g1, z4, z4, z8, 0);
#else
    __builtin_amdgcn_tensor_load_to_lds(g0, g1, z4, z4, 0);
#endif
}

// ---------------------------------------------------------------------------
// Kernel 3: out[b] = x[b] @ W[b]^T + ab[b]   via V_WMMA_F32_16X16X4_F32
// Block = 256 threads (8 wave32s). Block tile: 64 (M) x 256 (N = full D).
// Wave grid 2x4, wave tile 32x64 = 2x4 accumulators of 16x16.
// K chunked by 32; both tiles staged by the Tensor Data Mover.
// B fragment (k,n)=W[n][k]: contiguous along K in o-major W => aligned b64
// LDS reads, no transpose needed. Row stride 36 dwords: halves hit
// 0-mod-4 / 2-mod-4 bank pairs => conflict-free for all 32 lanes.
// ---------------------------------------------------------------------------
#define KC   32
#define TPAD 36

__global__ void token_mix_kernel(const float* __restrict__ x,   // [B,S,D]
                                 const float* __restrict__ Wws, // [B,D,D] (o-major)
                                 const float* __restrict__ abws,// [B,D]
                                 float* __restrict__ out)       // [B,S,D]
{
    __shared__ float s_x[64 * TPAD];    //  9216 B
    __shared__ float s_w[DIM * TPAD];   // 36864 B

    const int t    = threadIdx.x;
    const int lane = t & 31;
    const int wave = t >> 5;            // 0..7
    const int wm   = wave & 1;          // 2 waves along M (32 rows each)
    const int wn   = wave >> 1;         // 4 waves along N (64 cols each)

    const int b  = blockIdx.y;
    const int m0 = blockIdx.x * 64;     // S/64 = 8

    const float* xb = x   + (size_t)b * SEQ * DIM;
    const float* Wb = Wws + (size_t)b * DIM * DIM;

    v8f acc[2][4];
#pragma unroll
    for (int i = 0; i < 2; ++i)
#pragma unroll
        for (int j = 0; j < 4; ++j) acc[i][j] = (v8f){};

    const int half = (lane < 16) ? 0 : 2;   // K sub-offset for this lane half
    const int l16  = lane & 15;

    const unsigned lds_x = (unsigned)(uintptr_t)&s_x[0];
    const unsigned lds_w = (unsigned)(uintptr_t)&s_w[0];

    for (int k0 = 0; k0 < DIM; k0 += KC) {
        // ---- TDM: stage x (64 x 32) and W (256 x 32) chunks into padded LDS
        if (wave == 0) {
            tdm_load_tile_2d(lds_x, xb + (size_t)m0 * DIM + k0, KC, 64, DIM);
            tdm_load_tile_2d(lds_w, Wb + k0, KC, DIM, DIM);
            __builtin_amdgcn_s_wait_tensorcnt(0);
        }
        __syncthreads();

        // ---- 8 k-steps, 8 WMMAs each (2 M-tiles x 4 N-tiles)
#pragma unroll
        for (int k = 0; k < KC; k += 4) {
            v2f a[2], bf[4];
#pragma unroll
            for (int mt = 0; mt < 2; ++mt) {
                const float* p = &s_x[(wm * 32 + mt * 16 + l16) * TPAD + k + half];
                a[mt] = (v2f){ p[0], p[1] };
            }
#pragma unroll
            for (int nt = 0; nt < 4; ++nt) {
                const float* p = &s_w[(wn * 64 + nt * 16 + l16) * TPAD + k + half];
                bf[nt] = (v2f){ p[0], p[1] };
            }
#pragma unroll
            for (int mt = 0; mt < 2; ++mt)
#pragma unroll
                for (int nt = 0; nt < 4; ++nt)
                    acc[mt][nt] = __builtin_amdgcn_wmma_f32_16x16x4_f32(
                        false, a[mt], false, bf[nt], (short)0, acc[mt][nt],
                        false, false);
        }
        __syncthreads();
    }

    // ---- epilogue: fuse adaptive bias; D layout: lanes 0-15 -> rows r,
    // lanes 16-31 -> rows r+8, col = 16-lane index
    const int rofs = (lane < 16) ? 0 : 8;
    float* outb = out + (size_t)b * SEQ * DIM;

#pragma unroll
    for (int nt = 0; nt < 4; ++nt) {
        const int   ng  = wn * 64 + nt * 16 + l16;
        const float abv = abws[b * DIM + ng];
#pragma unroll
        for (int mt = 0; mt < 2; ++mt) {
            const int mbase = m0 + wm * 32 + mt * 16 + rofs;
#pragma unroll
            for (int r = 0; r < 8; ++r)
                outb[(size_t)(mbase + r) * DIM + ng] = acc[mt][nt][r] + abv;
        }
    }
}

// ---------------------------------------------------------------------------
extern "C" void kernel_launch(void* const* d_in, const int* in_sizes, int n_in,
                              void* d_out, int out_size, void* d_ws, size_t ws_size,
                              hipStream_t stream) {
    const float* x      = (const float*)d_in[0];
    const float* adapt  = (const float*)d_in[1];
    const float* weight = (const float*)d_in[2];
    const float* aw_w   = (const float*)d_in[3];
    const float* aw_b   = (const float*)d_in[4];
    const float* ab_w   = (const float*)d_in[5];
    const float* ab_b   = (const float*)d_in[6];
    float* out = (float*)d_out;

    float* Wws  = (float*)d_ws;                         // [B, D*D] = 1 MB
    float* abws = Wws + (size_t)BSZ * DIM * DIM;        // [B, D]   = 4 KB

    build_W_kernel<<<DIM * DIM / 256, 256, 0, stream>>>(adapt, weight, aw_w, aw_b, Wws);
    build_ab_kernel<<<1, 256, 0, stream>>>(adapt, ab_w, ab_b, abws);

    dim3 grid(SEQ / 64, BSZ);
    token_mix_kernel<<<grid, 256, 0, stream>>>(x, Wws, abws, out);
}